// HSSBlock_10368051052891
// MI455X (gfx1250) — compile-verified
//
#include <hip/hip_runtime.h>
#include <hip/hip_bf16.h>

// ---------------------------------------------------------------------------
// VMamba SS2D block for MI455X (gfx1250, wave32).
//   B=8, C=96, H=W=64, D=192, N=16, R=6, K=4, L=4096
// Dense projections use V_WMMA_F32_16X16X4_F32 (fp32 WMMA, no downcast).
// Selective scan: one thread per (b,k,d), N=16 state in registers, B/C/dt
// chunks staged in LDS, results merged by spatial position with
// global_atomic_add_f32 (all 4 directions land on row-major positions).
// ---------------------------------------------------------------------------

#define BB   8
#define CC   96
#define HH   64
#define WWW  64
#define HW   4096
#define DD   192
#define KK   4
#define NN   16
#define RR   6
#define LLn  4096
#define EPS  1e-5f

typedef __attribute__((ext_vector_type(2))) float v2f;
typedef __attribute__((ext_vector_type(8))) float v8f;

// direction index -> row-major spatial position (H=W=64, L=4096)
__device__ __forceinline__ int posOf(int k, int l) {
    switch (k) {
        case 0:  return l;
        case 1:  return ((l & 63) << 6) | (l >> 6);
        case 2:  return 4095 - l;
        default: { int lr = 4095 - l; return ((lr & 63) << 6) | (lr >> 6); }
    }
}

// ---------------------------------------------------------------------------
// K0: fold BatchNorm into in_proj:  Wf[e,c] = W[e,c]*s[c],  bf[e] = sum W[e,c]*t[c]
// ---------------------------------------------------------------------------
__global__ void k_fold_bn(const float* __restrict__ W,
                          const float* __restrict__ gamma, const float* __restrict__ beta,
                          const float* __restrict__ mean,  const float* __restrict__ var,
                          float* __restrict__ Wf, float* __restrict__ bf) {
    int e = blockIdx.x * blockDim.x + threadIdx.x;
    if (e >= 2 * DD) return;
    float acc = 0.f;
    for (int c = 0; c < CC; ++c) {
        float s = gamma[c] * rsqrtf(var[c] + EPS);
        float t = beta[c] - mean[c] * s;
        float w = W[e * CC + c];
        Wf[e * CC + c] = w * s;
        acc += w * t;
    }
    bf[e] = acc;
}

// ---------------------------------------------------------------------------
// K1: in_proj GEMM  xz[m,e] = sum_c Wf[e,c]*x[b,c,p] + bf[e]
//     m = b*4096+p (32768 rows), e in [0,384). e<192 -> xc (B,D,HW) for conv,
//     e>=192 -> z (B,HW,D) for gating.  fp32 WMMA 16x16x4, K-loop 96/4=24.
// ---------------------------------------------------------------------------
__global__ void k_in_proj(const float* __restrict__ x, const float* __restrict__ Wf,
                          const float* __restrict__ bf,
                          float* __restrict__ xc, float* __restrict__ z) {
    const int wave = (blockIdx.x * blockDim.x + threadIdx.x) >> 5;
    const int lane = threadIdx.x & 31;
    const int nM = (BB * HW) / 16;   // 2048
    const int nE = (2 * DD) / 16;    // 24
    if (wave >= nM * nE) return;     // wave-uniform
    const int mt = wave / nE, et = wave % nE;

    const int row   = mt * 16 + (lane & 15);
    const int b     = row >> 12, p = row & 4095;
    const int e     = et * 16 + (lane & 15);
    const int khalf = (lane >> 4) * 2;

    const float* xb = x + (b * CC) * HW + p;
    v8f c = {};
    for (int k0 = 0; k0 < CC; k0 += 4) {
        const int ks = k0 + khalf;
        v2f a; a[0] = xb[ks * HW];       a[1] = xb[(ks + 1) * HW];
        v2f w; w[0] = Wf[e * CC + ks];   w[1] = Wf[e * CC + ks + 1];
        c = __builtin_amdgcn_wmma_f32_16x16x4_f32(false, a, false, w, (short)0, c, false, false);
    }
    const float bias  = bf[e];
    const int   rbase = mt * 16 + ((lane >> 4) << 3);
#pragma unroll
    for (int j = 0; j < 8; ++j) {
        const float v = c[j] + bias;
        const int r = rbase + j, bb = r >> 12, pp = r & 4095;
        if (e < DD) xc[(bb * DD + e) * HW + pp]      = v;  // conv input
        else        z[(bb * HW + pp) * DD + (e-DD)]  = v;  // gate branch
    }
}

// ---------------------------------------------------------------------------
// K2: depthwise 3x3 conv (pad 1) + bias + SiLU -> u (B,D,HW)
// ---------------------------------------------------------------------------
__global__ void k_conv_silu(const float* __restrict__ xc, const float* __restrict__ cw,
                            const float* __restrict__ cb, float* __restrict__ u) {
    const int bd = blockIdx.x;           // B*D planes
    const int b = bd / DD, d = bd % DD;
    const float* src = xc + (b * DD + d) * HW;
    float w[9];
#pragma unroll
    for (int i = 0; i < 9; ++i) w[i] = cw[d * 9 + i];
    const float bias = cb[d];
    for (int p = threadIdx.x; p < HW; p += blockDim.x) {
        const int h = p >> 6, ww = p & 63;
        float acc = bias;
#pragma unroll
        for (int dy = -1; dy <= 1; ++dy)
#pragma unroll
            for (int dx = -1; dx <= 1; ++dx) {
                const int hh = h + dy, wc = ww + dx;
                if (hh >= 0 && hh < HH && wc >= 0 && wc < WWW)
                    acc += w[(dy + 1) * 3 + (dx + 1)] * src[(hh << 6) | wc];
            }
        u[(b * DD + d) * HW + p] = acc / (1.f + expf(-acc)); // SiLU
    }
}

// ---------------------------------------------------------------------------
// K3: x_proj GEMM per (b,k):  proj[l,col] = sum_d xpw[k,col,d] * u[b,d,pos(k,l)]
//     38 cols padded to 48 (3 tiles). col<6 -> dtr, <22 -> Bs, <38 -> Cs.
// ---------------------------------------------------------------------------
__global__ void k_xproj(const float* __restrict__ u, const float* __restrict__ xpw,
                        float* __restrict__ dtr, float* __restrict__ Bs,
                        float* __restrict__ Cs) {
    const int wave = (blockIdx.x * blockDim.x + threadIdx.x) >> 5;
    const int lane = threadIdx.x & 31;
    const int nLt = LLn / 16;            // 256
    const int tilesPerBK = nLt * 3;      // 768
    if (wave >= BB * KK * tilesPerBK) return;  // wave-uniform
    const int bk = wave / tilesPerBK, rem = wave % tilesPerBK;
    const int lt = rem / 3, ct = rem % 3;
    const int b = bk >> 2, k = bk & 3;

    const int l     = lt * 16 + (lane & 15);
    const int p     = posOf(k, l);
    const int col   = ct * 16 + (lane & 15);
    const int khalf = (lane >> 4) * 2;
    const bool colValid = col < (RR + 2 * NN);

    const float* ub = u + (b * DD) * HW + p;
    const float* wb = xpw + k * (RR + 2 * NN) * DD;
    v8f c = {};
    for (int k0 = 0; k0 < DD; k0 += 4) {
        const int ks = k0 + khalf;
        v2f a; a[0] = ub[ks * HW]; a[1] = ub[(ks + 1) * HW];
        v2f w;
        if (colValid) { w[0] = wb[col * DD + ks]; w[1] = wb[col * DD + ks + 1]; }
        else          { w[0] = 0.f;               w[1] = 0.f; }
        c = __builtin_amdgcn_wmma_f32_16x16x4_f32(false, a, false, w, (short)0, c, false, false);
    }
    const int  lbase = lt * 16 + ((lane >> 4) << 3);
    const long bkL   = (long)bk * LLn;
#pragma unroll
    for (int j = 0; j < 8; ++j) {
        const int  ll = lbase + j;
        const float v = c[j];
        if      (col < RR)          dtr[(bkL + ll) * RR + col]            = v;
        else if (col < RR + NN)     Bs [(bkL + ll) * NN + (col - RR)]     = v;
        else if (col < RR + 2*NN)   Cs [(bkL + ll) * NN + (col - RR - NN)] = v;
    }
}

// ---------------------------------------------------------------------------
// K4: selective scan. Block = (b,k), thread = d. h[16] in registers.
//     dt = softplus(dt_w[k,d,:]·dtr + dt_b); h = e^{dt*A}h + dt*u*B; y = h·C.
//     Adds (y + Ds*u) into merged[b,p,d] with float atomics (4 dirs merge).
// ---------------------------------------------------------------------------
#define CHUNK 64
__global__ void k_scan(const float* __restrict__ u,   const float* __restrict__ dtr,
                       const float* __restrict__ Bs,  const float* __restrict__ Cs,
                       const float* __restrict__ dtw, const float* __restrict__ dtb,
                       const float* __restrict__ Alogs, const float* __restrict__ Dsk,
                       float* __restrict__ merged) {
    __shared__ float sB[CHUNK * NN], sC[CHUNK * NN], sR[CHUNK * RR];
    const int bk = blockIdx.x, b = bk >> 2, k = bk & 3;
    const int d  = threadIdx.x;          // 0..191
    const int kd = k * DD + d;

    float a[NN], h[NN];
#pragma unroll
    for (int n = 0; n < NN; ++n) { a[n] = -expf(Alogs[kd * NN + n]); h[n] = 0.f; }
    float wr[RR];
#pragma unroll
    for (int r = 0; r < RR; ++r) wr[r] = dtw[kd * RR + r];
    const float bdt = dtb[kd], dskip = Dsk[kd];
    const float* ubd = u + (b * DD + d) * HW;
    const long bkL = (long)bk * LLn;

    for (int l0 = 0; l0 < LLn; l0 += CHUNK) {
        __syncthreads();
        for (int i = threadIdx.x; i < CHUNK * NN; i += blockDim.x) {
            sB[i] = Bs[(bkL + l0) * NN + i];
            sC[i] = Cs[(bkL + l0) * NN + i];
        }
        for (int i = threadIdx.x; i < CHUNK * RR; i += blockDim.x)
            sR[i] = dtr[(bkL + l0) * RR + i];
        __syncthreads();
        for (int li = 0; li < CHUNK; ++li) {
            const int l = l0 + li;
            float acc = bdt;
#pragma unroll
            for (int r = 0; r < RR; ++r) acc += sR[li * RR + r] * wr[r];
            const float dt = acc > 20.f ? acc : log1pf(expf(acc)); // softplus
            const int   p  = posOf(k, l);
            const float uv = ubd[p];
            const float du = dt * uv;
            float y = 0.f;
#pragma unroll
            for (int n = 0; n < NN; ++n) {
                h[n] = expf(dt * a[n]) * h[n] + du * sB[li * NN + n];
                y += h[n] * sC[li * NN + n];
            }
            atomicAdd(&merged[((long)b * HW + p) * DD + d], y + dskip * uv);
        }
    }
}

// ---------------------------------------------------------------------------
// K5: LayerNorm over D + SiLU gate:  g = LN(merged) * silu(z)
// ---------------------------------------------------------------------------
__global__ void k_ln_gate(const float* __restrict__ merged, const float* __restrict__ z,
                          const float* __restrict__ lnw, const float* __restrict__ lnb,
                          float* __restrict__ g) {
    __shared__ float red[12];
    __shared__ float stats[2];
    const int bp = blockIdx.x, d = threadIdx.x;   // 192 threads = 6 waves
    const long base = (long)bp * DD;
    const float v = merged[base + d];
    float s = v, s2 = v * v;
#pragma unroll
    for (int m = 16; m >= 1; m >>= 1) { s += __shfl_xor(s, m, 32); s2 += __shfl_xor(s2, m, 32); }
    if ((d & 31) == 0) { red[d >> 5] = s; red[6 + (d >> 5)] = s2; }
    __syncthreads();
    if (d == 0) {
        float ts = 0.f, ts2 = 0.f;
        for (int i = 0; i < 6; ++i) { ts += red[i]; ts2 += red[6 + i]; }
        const float mu = ts / DD;
        stats[0] = mu; stats[1] = rsqrtf(ts2 / DD - mu * mu + EPS);
    }
    __syncthreads();
    const float yn = (v - stats[0]) * stats[1] * lnw[d] + lnb[d];
    const float zv = z[base + d];
    g[base + d] = yn * (zv / (1.f + expf(-zv)));
}

// ---------------------------------------------------------------------------
// K6: out_proj GEMM + residual:  out[b,c,p] = x[b,c,p] + sum_d Wo[c,d]*g[b,p,d]
// ---------------------------------------------------------------------------
__global__ void k_out_proj(const float* __restrict__ g, const float* __restrict__ Wo,
                           const float* __restrict__ x, float* __restrict__ out) {
    const int wave = (blockIdx.x * blockDim.x + threadIdx.x) >> 5;
    const int lane = threadIdx.x & 31;
    const int nM = (BB * HW) / 16;   // 2048
    const int nC = CC / 16;          // 6
    if (wave >= nM * nC) return;     // wave-uniform
    const int mt = wave / nC, ct = wave % nC;
    const int row   = mt * 16 + (lane & 15);
    const int cidx  = ct * 16 + (lane & 15);
    const int khalf = (lane >> 4) * 2;

    const float* gr = g + (long)row * DD;
    v8f c = {};
    for (int k0 = 0; k0 < DD; k0 += 4) {
        const int ks = k0 + khalf;
        v2f a; a[0] = gr[ks];             a[1] = gr[ks + 1];
        v2f w; w[0] = Wo[cidx * DD + ks]; w[1] = Wo[cidx * DD + ks + 1];
        c = __builtin_amdgcn_wmma_f32_16x16x4_f32(false, a, false, w, (short)0, c, false, false);
    }
    const int rbase = mt * 16 + ((lane >> 4) << 3);
#pragma unroll
    for (int j = 0; j < 8; ++j) {
        const int r = rbase + j, b = r >> 12, p = r & 4095;
        const long idx = ((long)b * CC + cidx) * HW + p;
        out[idx] = x[idx] + c[j];
    }
}

// ---------------------------------------------------------------------------
extern "C" void kernel_launch(void* const* d_in, const int* in_sizes, int n_in,
                              void* d_out, int out_size, void* d_ws, size_t ws_size,
                              hipStream_t stream) {
    const float* x      = (const float*)d_in[0];
    const float* bng    = (const float*)d_in[1];
    const float* bnb    = (const float*)d_in[2];
    const float* bnm    = (const float*)d_in[3];
    const float* bnv    = (const float*)d_in[4];
    const float* inW    = (const float*)d_in[5];
    const float* convW  = (const float*)d_in[6];
    const float* convB  = (const float*)d_in[7];
    const float* xprojW = (const float*)d_in[8];
    const float* dtW    = (const float*)d_in[9];
    const float* dtB    = (const float*)d_in[10];
    const float* Alogs  = (const float*)d_in[11];
    const float* Dsk    = (const float*)d_in[12];
    const float* lnW    = (const float*)d_in[13];
    const float* lnB    = (const float*)d_in[14];
    const float* outW   = (const float*)d_in[15];
    float* out = (float*)d_out;

    // workspace layout (bytes)
    char* w = (char*)d_ws;
    const size_t SZ_BDHW = (size_t)BB * DD * HW * 4;      // 25165824
    float* xc     = (float*)(w);                          // pre-conv; reused as merged
    float* z      = (float*)(w + SZ_BDHW);
    float* u      = (float*)(w + 2 * SZ_BDHW);
    char*  w3     = w + 3 * SZ_BDHW;
    float* dtr    = (float*)(w3);                                         // B*K*L*R
    float* Bs     = (float*)(w3 + (size_t)BB*KK*LLn*RR*4);                // B*K*L*N
    float* Cs     = (float*)(w3 + (size_t)BB*KK*LLn*RR*4 + (size_t)BB*KK*LLn*NN*4);
    char*  w4     = w3 + (size_t)BB*KK*LLn*(RR + 2*NN)*4;
    float* g      = (float*)(w4);                          // gated (B,HW,D)
    float* Wf     = (float*)(w4 + SZ_BDHW);
    float* bf     = (float*)(w4 + SZ_BDHW + (size_t)2*DD*CC*4);
    float* merged = xc;                                    // reuse after conv

    // K0: fold BN into in_proj
    k_fold_bn<<<(2 * DD + 255) / 256, 256, 0, stream>>>(inW, bng, bnb, bnm, bnv, Wf, bf);

    // K1: in_proj WMMA GEMM (2048*24 tiles, 8 waves/block)
    k_in_proj<<<(2048 * 24) / 8, 256, 0, stream>>>(x, Wf, bf, xc, z);

    // K2: depthwise conv + SiLU
    k_conv_silu<<<BB * DD, 256, 0, stream>>>(xc, convW, convB, u);

    // zero the merge accumulator (reuses xc region; conv already consumed it)
    hipMemsetAsync(merged, 0, SZ_BDHW, stream);

    // K3: x_proj WMMA GEMM (32 bk * 768 tiles)
    k_xproj<<<(BB * KK * 768) / 8, 256, 0, stream>>>(u, xprojW, dtr, Bs, Cs);

    // K4: selective scan + directional merge via atomics
    k_scan<<<BB * KK, DD, 0, stream>>>(u, dtr, Bs, Cs, dtW, dtB, Alogs, Dsk, merged);

    // K5: LayerNorm + SiLU gate
    k_ln_gate<<<BB * HW, DD, 0, stream>>>(merged, z, lnW, lnB, g);

    // K6: out_proj WMMA GEMM + residual
    k_out_proj<<<(2048 * 6) / 8, 256, 0, stream>>>(g, outW, x, out);
}